// Equilibrium_23579370455609
// MI455X (gfx1250) — compile-verified
//
#include <hip/hip_runtime.h>
#include <hip/hip_bf16.h>
#include <stddef.h>
#include <stdint.h>

// Shapes from the reference
#define T_DIM 16
#define M_DIM 1024
#define N_NODES 10000
#define K_DIM 4096          // 4*M
#define N_COLS 20000        // 2*N_NODES
#define N_IN 8000
#define N_B 100
#define KSEG 512            // K per wave (8 waves split K=4096)
#define CHUNK_K 64          // K values staged per async chunk
#define NODE_STRIDE 136     // floats per node row in LDS (128 used + 8 pad -> conflict-free)
#define BUF_FLOATS (8 * NODE_STRIDE)   // 1088 floats per chunk buffer

typedef float v2f __attribute__((ext_vector_type(2)));
typedef float v8f __attribute__((ext_vector_type(8)));

// ---------------------------------------------------------------------------
// Kernel 1: build x_out [16][4096]:  x_out[t, 4m+r] = x[t, ch(r), m],
// ch = {0,2,2,1}
// ---------------------------------------------------------------------------
__global__ void build_xout_kernel(const float* __restrict__ x,
                                  float* __restrict__ xout) {
    int i = blockIdx.x * blockDim.x + threadIdx.x;
    if (i >= T_DIM * K_DIM) return;
    int t = i >> 12;          // / 4096
    int c = i & (K_DIM - 1);
    int m = c >> 2;
    int r = c & 3;
    int ch = (r == 0) ? 0 : ((r == 3) ? 1 : 2);
    xout[i] = x[t * (3 * M_DIM) + ch * M_DIM + m];
}

// ---------------------------------------------------------------------------
// Kernel 2: GEMM  f[n,t,o] = sum_k xout[t,k] * weight[n,k,o]
// via V_WMMA_F32_16X16X4_F32, with weight streamed into LDS using the CDNA5
// async copy path (GLOBAL_LOAD_ASYNC_TO_LDS_B128 + ASYNCcnt), double-buffered
// per wave.  One block = one 16-column tile (8 nodes); 8 waves split K.
// ---------------------------------------------------------------------------
__global__ void __launch_bounds__(256)
equilibrium_gemm_kernel(const float* __restrict__ xout,
                        const float* __restrict__ weight,
                        float* __restrict__ f) {
    __shared__ float bstage[8][2][BUF_FLOATS];   // 68 KB: per-wave double buffer
    __shared__ float red[8 * 256];               // 8 KB: split-K reduction

    const int tile = blockIdx.x;          // 0..1249
    const int tid  = threadIdx.x;         // 0..255
    const int wave = tid >> 5;            // 0..7  -> K segment
    const int lane = tid & 31;
    const int N    = lane & 15;           // column within tile / A row (M)
    const int half = lane >> 4;           // 0: K=0,1   1: K=2,3

    const int col  = tile * 16 + N;       // 0..19999
    const int node = col >> 1;            // (unused for staging; per-lane frag)
    const int o    = col & 1;
    const int nl   = N >> 1;              // node-local index within tile

    // A fragment source: xout[M = lane&15][k + half*2 + {0,1}] -> float2 load
    const float* __restrict__ aptr = xout + (size_t)N * K_DIM + (half << 1);

    // Weight base for this tile (8 nodes, each row 8192 floats)
    const float* __restrict__ wbase = weight + (size_t)(tile * 8) * (K_DIM * 2);

    // LDS base (32-bit LDS byte offset = low 32 bits of the generic address)
    const uint32_t lds_base =
        (uint32_t)(uintptr_t)(&bstage[wave][0][0]);

    const int kbeg = wave * KSEG;

    // ---- async stage: chunk = 8 nodes x 64 K x 2 outs (4 KB), one b128/node
    auto issue_chunk = [&](int bufsel, int k) {
#pragma unroll
        for (int j = 0; j < 8; ++j) {
            const float* g = wbase + (size_t)j * (K_DIM * 2)
                                   + (size_t)k * 2 + (size_t)lane * 4;
            uint32_t l = lds_base
                       + (uint32_t)(bufsel * BUF_FLOATS + j * NODE_STRIDE) * 4u
                       + (uint32_t)lane * 16u;
            asm volatile("global_load_async_to_lds_b128 %0, %1, off"
                         :: "v"(l), "v"(g) : "memory");
        }
    };

    v8f acc0 = {};
    v8f acc1 = {};

    issue_chunk(0, kbeg);                 // prologue: fill buffer 0

    int bufsel = 0;
    for (int c = 0; c < KSEG / CHUNK_K; ++c) {     // 8 chunks (uniform branch)
        if (c < KSEG / CHUNK_K - 1) {
            issue_chunk(bufsel ^ 1, kbeg + (c + 1) * CHUNK_K);
            asm volatile("s_wait_asynccnt 0x8" ::: "memory");
        } else {
            asm volatile("s_wait_asynccnt 0x0" ::: "memory");
        }

        const float* buf = &bstage[wave][bufsel][0];
        const int kbase = kbeg + c * CHUNK_K;
        const int fb = nl * NODE_STRIDE + (half << 1) * 2 + o;  // frag base

#pragma unroll
        for (int kk = 0; kk < CHUNK_K; kk += 8) {
            v2f a0 = *(const v2f*)(aptr + kbase + kk);
            v2f a1 = *(const v2f*)(aptr + kbase + kk + 4);
            int i0 = fb + kk * 2;
            int i1 = fb + (kk + 4) * 2;
            v2f b0, b1;
            b0.x = buf[i0];
            b0.y = buf[i0 + 2];
            b1.x = buf[i1];
            b1.y = buf[i1 + 2];
            acc0 = __builtin_amdgcn_wmma_f32_16x16x4_f32(
                false, a0, false, b0, (short)0, acc0, false, false);
            acc1 = __builtin_amdgcn_wmma_f32_16x16x4_f32(
                false, a1, false, b1, (short)0, acc1, false, false);
        }
        bufsel ^= 1;
    }

    v8f acc = acc0 + acc1;

    // Split-K reduction through LDS: partial[wave][t*16 + N]
    float* my = red + wave * 256;
#pragma unroll
    for (int v = 0; v < 8; ++v) {
        int t = v + (half << 3);          // C layout: M = v + 8*(lane/16)
        my[t * 16 + N] = acc[v];
    }
    __syncthreads();

    // One thread per tile element: sum 8 partials, store to f[n][t][o]
    float s = 0.f;
#pragma unroll
    for (int w = 0; w < 8; ++w) s += red[w * 256 + tid];
    int t2 = tid >> 4;
    int N2 = tid & 15;
    int col2 = tile * 16 + N2;
    int n2 = col2 >> 1;
    int o2 = col2 & 1;
    f[(size_t)n2 * (2 * T_DIM) + t2 * 2 + o2] = s;
    (void)node;
}

// ---------------------------------------------------------------------------
// Kernel 3: f_in gather  (8000 x 16 x 2), contiguous 32-float rows
// ---------------------------------------------------------------------------
__global__ void gather_fin_kernel(const float* __restrict__ f,
                                  const int* __restrict__ node_in,
                                  float* __restrict__ out) {
    int i = blockIdx.x * blockDim.x + threadIdx.x;
    if (i >= N_IN * 32) return;
    int j = i >> 5;
    int q = i & 31;
    out[i] = f[(size_t)(node_in[j] - 1) * 32 + q];
}

// ---------------------------------------------------------------------------
// Kernel 4: boundary sums f_b [6][16][1]
// ---------------------------------------------------------------------------
__global__ void compute_fb_kernel(const float* __restrict__ f,
                                  const int* __restrict__ top,
                                  const int* __restrict__ bot,
                                  const int* __restrict__ lef,
                                  const int* __restrict__ rig,
                                  float* __restrict__ out) {
    int idx = threadIdx.x;
    if (idx >= 96) return;
    int r = idx >> 4;
    int t = idx & 15;
    auto rowsum = [&](const int* a, int oo) {
        float s = 0.f;
        for (int i = 0; i < N_B; ++i)
            s += f[(size_t)(a[i] - 1) * 32 + t * 2 + oo];
        return s;
    };
    float val;
    switch (r) {
        case 0:  val = rowsum(top, 1); break;
        case 1:  val = rowsum(bot, 1); break;
        case 2:  val = rowsum(lef, 0); break;
        case 3:  val = rowsum(rig, 0); break;
        case 4:  val = rowsum(top, 1) + rowsum(bot, 1); break;
        default: val = rowsum(lef, 0) + rowsum(rig, 0); break;
    }
    out[N_IN * 32 + idx] = val;
}

// ---------------------------------------------------------------------------
extern "C" void kernel_launch(void* const* d_in, const int* in_sizes, int n_in,
                              void* d_out, int out_size, void* d_ws, size_t ws_size,
                              hipStream_t stream) {
    const float* x       = (const float*)d_in[0];
    const float* weight  = (const float*)d_in[1];
    const int*   node_in = (const int*)d_in[2];
    const int*   top     = (const int*)d_in[3];
    const int*   bottom  = (const int*)d_in[4];
    const int*   left    = (const int*)d_in[5];
    const int*   right   = (const int*)d_in[6];
    float*       out     = (float*)d_out;

    // Workspace: xout [16*4096] floats, then f [10000*32] floats
    float* xout = (float*)d_ws;
    float* f    = xout + (size_t)T_DIM * K_DIM;

    // 1) build x_out
    build_xout_kernel<<<(T_DIM * K_DIM + 255) / 256, 256, 0, stream>>>(x, xout);

    // 2) GEMM: 1250 tiles of 16 columns, 8 split-K waves per block
    equilibrium_gemm_kernel<<<N_COLS / 16, 256, 0, stream>>>(xout, weight, f);

    // 3) gather f_in
    gather_fin_kernel<<<(N_IN * 32 + 255) / 256, 256, 0, stream>>>(f, node_in, out);

    // 4) boundary sums
    compute_fb_kernel<<<1, 128, 0, stream>>>(f, top, bottom, left, right, out);
}